// Spatio_Temporal_Mutual_Attention_27599459844724
// MI455X (gfx1250) — compile-verified
//
#include <hip/hip_runtime.h>
#include <math.h>

// ---------------------------------------------------------------------------
// Spatio-temporal mutual (channel) attention, fused for MI455X / gfx1250.
//   inputs : inp1[8,192,128,128] inp2[..] Wq/bq Wk/bk Wv/bv temperature[4]
//   outputs: attn[8,4,48,48] ++ v[8,192,128,128]   (concatenated in d_out)
// ---------------------------------------------------------------------------

typedef __attribute__((ext_vector_type(2))) float v2f;
typedef __attribute__((ext_vector_type(8))) float v8f;

#define CCH    192      // channels
#define TILE_P 128      // pixels per workgroup tile
#define HW     16384    // 128*128
#define NHEAD  4
#define CPH    48       // channels per head
#define ATTN_N (8 * NHEAD * CPH * CPH)   // 73728

#if __has_builtin(__builtin_amdgcn_wmma_f32_16x16x4_f32)
#define HAVE_WMMA_F32_4 1
#else
#define HAVE_WMMA_F32_4 0
#endif

__device__ __forceinline__ v8f wmma_f32(v2f a, v2f b, v8f c) {
#if HAVE_WMMA_F32_4
    // 8 args: (neg_a, A, neg_b, B, c_mod, C, reuse_a, reuse_b)
    return __builtin_amdgcn_wmma_f32_16x16x4_f32(false, a, false, b,
                                                 (short)0, c, false, false);
#else
#pragma unroll
    for (int r = 0; r < 8; ++r) c[r] += a.x * b.x + a.y * b.y;
    return c;
#endif
}

// Stage a [192 x 128] f32 tile (global, row stride HW) into LDS via the
// gfx1250 async global->LDS path: direct to LDS, no VGPR round trip,
// tracked by ASYNCcnt instead of LOADcnt.
__device__ __forceinline__ void async_tile_load(const float* __restrict__ src,
                                                float* dstLds, int tid) {
    for (int idx = tid; idx < CCH * (TILE_P / 4); idx += 256) {
        const int c = idx >> 5, p4 = idx & 31;
        const float* g = src + (size_t)c * HW + p4 * 4;
        unsigned lds = (unsigned)(size_t)(dstLds + c * TILE_P + p4 * 4);
        asm volatile("global_load_async_to_lds_b128 %0, %1, off"
                     :
                     : "v"(lds), "v"(g)
                     : "memory");
    }
    asm volatile("s_wait_asynccnt 0x0" ::: "memory");
}

// One 16(row)x16(pixel) output tile of  W[192x192] * Xs[192x128] + bias.
// A-frag layout (f32 16x16x4): lane<16 holds K=koff..koff+1 of row M=lane&15.
__device__ __forceinline__ v8f conv_tile(const float* __restrict__ W,
                                         const float* __restrict__ bias,
                                         const float* __restrict__ Xs,
                                         int m, int l15, int half, int koff,
                                         int px) {
    const int arow = 16 * m + l15;
    v8f acc;
#pragma unroll
    for (int r = 0; r < 8; ++r) acc[r] = bias[16 * m + r + 8 * half];
#pragma unroll 4
    for (int kk = 0; kk < CCH / 4; ++kk) {
        const int c0 = 4 * kk;
        v2f a = *(const v2f*)(W + (size_t)arow * CCH + c0 + koff); // 8B aligned
        v2f b;
        b.x = Xs[(c0 + koff) * TILE_P + px];
        b.y = Xs[(c0 + koff + 1) * TILE_P + px];
        acc = wmma_f32(a, b, acc);
    }
    return acc;
}

__global__ void __launch_bounds__(256) stmua_zero_kernel(float* p, int n) {
    int i = blockIdx.x * 256 + threadIdx.x;
    if (i < n) p[i] = 0.f;
}

__global__ void __launch_bounds__(256) stmua_stage1(
    const float* __restrict__ x1, const float* __restrict__ x2,
    const float* __restrict__ Wq, const float* __restrict__ bq,
    const float* __restrict__ Wk, const float* __restrict__ bk,
    const float* __restrict__ Wv, const float* __restrict__ bv,
    float* __restrict__ vOut,  // d_out + ATTN_N
    float* __restrict__ sAcc,  // [8,4,48,48] gram accumulator (ws)
    float* __restrict__ qss,   // [8,192] sum-of-squares q rows (ws)
    float* __restrict__ kss)   // [8,192] sum-of-squares k rows (ws)
{
    extern __shared__ float smem[];
    float* Xs = smem;                      // 192*128 f32 (inp1 then inp2)
    float* Ks = smem + CCH * TILE_P;       // k tile
    float* Qs = Ks + CCH * TILE_P;         // q tile

    const int tid  = threadIdx.x;
    const int lane = tid & 31;
    const int wave = tid >> 5;
    const int l15  = lane & 15;
    const int half = lane >> 4;
    const int koff = half * 2;
    const int px   = wave * 16 + l15;      // this lane's pixel column

    const int    bi      = blockIdx.x >> 7;    // batch
    const int    t       = blockIdx.x & 127;   // pixel tile in batch
    const size_t p0      = (size_t)t * TILE_P;
    const size_t imgBase = (size_t)bi * CCH * HW;

    const float* src1 = x1 + imgBase + p0;
    const float* src2 = x2 + imgBase + p0;

    // ---- Phase 1: inp1 tile -> LDS (async DMA path) -----------------------
    async_tile_load(src1, Xs, tid);
    // prefetch inp2 tile into caches while phase-1 math runs (global_prefetch_b8)
    for (int c = tid; c < CCH; c += 256) {
        __builtin_prefetch(src2 + (size_t)c * HW, 0, 1);
        __builtin_prefetch(src2 + (size_t)c * HW + 64, 0, 1);
    }
    __syncthreads();

    // v = Wv*X1 + bv  -> straight to d_out ; k = Wk*X1 + bk -> LDS
#pragma unroll 1
    for (int m = 0; m < CCH / 16; ++m) {
        v8f acc = conv_tile(Wv, bv, Xs, m, l15, half, koff, px);
#pragma unroll
        for (int r = 0; r < 8; ++r)
            vOut[((size_t)bi * CCH + 16 * m + r + 8 * half) * HW + p0 + px] =
                acc[r];
    }
#pragma unroll 1
    for (int m = 0; m < CCH / 16; ++m) {
        v8f acc = conv_tile(Wk, bk, Xs, m, l15, half, koff, px);
#pragma unroll
        for (int r = 0; r < 8; ++r)
            Ks[(16 * m + r + 8 * half) * TILE_P + px] = acc[r];
    }
    __syncthreads();

    // ---- Phase 2: inp2 tile -> LDS (reuse Xs), q -> LDS -------------------
    async_tile_load(src2, Xs, tid);
    __syncthreads();
#pragma unroll 1
    for (int m = 0; m < CCH / 16; ++m) {
        v8f acc = conv_tile(Wq, bq, Xs, m, l15, half, koff, px);
#pragma unroll
        for (int r = 0; r < 8; ++r)
            Qs[(16 * m + r + 8 * half) * TILE_P + px] = acc[r];
    }
    __syncthreads();

    // ---- Phase 3: partial Gram  S_h += Q_h * K_h^T  (contract 128 px) -----
    // 4 heads * 3x3 row/col tiles = 36 16x16 tiles, round-robined over 8 waves
    for (int tIdx = wave; tIdx < NHEAD * 9; tIdx += 8) {
        const int h  = tIdx / 9;
        const int ti = (tIdx % 9) / 3;
        const int tj = tIdx % 3;
        v8f s = {0.f, 0.f, 0.f, 0.f, 0.f, 0.f, 0.f, 0.f};
        const float* qrow = Qs + (CPH * h + 16 * ti + l15) * TILE_P;
        const float* krow = Ks + (CPH * h + 16 * tj + l15) * TILE_P;
#pragma unroll 4
        for (int kk = 0; kk < TILE_P / 4; ++kk) {
            v2f a = *(const v2f*)(qrow + 4 * kk + koff);  // ds_load_b64
            v2f b = *(const v2f*)(krow + 4 * kk + koff);
            s = wmma_f32(a, b, s);
        }
        float* sBase = sAcc + ((size_t)bi * NHEAD + h) * CPH * CPH;
#pragma unroll
        for (int r = 0; r < 8; ++r) {
            const int row = 16 * ti + r + 8 * half;
            const int col = 16 * tj + l15;
            atomicAdd(&sBase[row * CPH + col], s[r]);
        }
    }

    // ---- Phase 3b: row sum-of-squares for the L2 norms --------------------
    if (tid < CCH) {
        float sq = 0.f, sk = 0.f;
        const float* qr = Qs + tid * TILE_P;
        const float* kr = Ks + tid * TILE_P;
#pragma unroll 8
        for (int p = 0; p < TILE_P; ++p) {
            const float q = qr[p]; sq += q * q;
            const float k = kr[p]; sk += k * k;
        }
        atomicAdd(&qss[bi * CCH + tid], sq);
        atomicAdd(&kss[bi * CCH + tid], sk);
    }
}

// ---- Stage 2: normalize by ||q||*||k||, * temperature, softmax ------------
__global__ void __launch_bounds__(64) stmua_softmax(
    const float* __restrict__ sAcc, const float* __restrict__ qss,
    const float* __restrict__ kss, const float* __restrict__ temp,
    float* __restrict__ attnOut)
{
    const int bh = blockIdx.x;
    const int b  = bh >> 2;
    const int h  = bh & 3;
    __shared__ float knS[CPH];
    const int tid = threadIdx.x;
    if (tid < CPH)
        knS[tid] = fmaxf(sqrtf(kss[b * CCH + h * CPH + tid]), 1e-12f);
    __syncthreads();
    if (tid < CPH) {
        const float qn = fmaxf(sqrtf(qss[b * CCH + h * CPH + tid]), 1e-12f);
        const float tmp = temp[h];
        const float* srow = sAcc + (((size_t)b * NHEAD + h) * CPH + tid) * CPH;
        float l[CPH];
        float mx = -INFINITY;
#pragma unroll
        for (int j = 0; j < CPH; ++j) {
            const float v = srow[j] / (qn * knS[j]) * tmp;
            l[j] = v;
            mx = fmaxf(mx, v);
        }
        float sum = 0.f;
#pragma unroll
        for (int j = 0; j < CPH; ++j) {
            const float e = __expf(l[j] - mx);
            l[j] = e;
            sum += e;
        }
        const float inv = 1.f / sum;
        float* orow = attnOut + (((size_t)b * NHEAD + h) * CPH + tid) * CPH;
#pragma unroll
        for (int j = 0; j < CPH; ++j) orow[j] = l[j] * inv;
    }
}

extern "C" void kernel_launch(void* const* d_in, const int* in_sizes, int n_in,
                              void* d_out, int out_size, void* d_ws,
                              size_t ws_size, hipStream_t stream) {
    const float* inp1 = (const float*)d_in[0];
    const float* inp2 = (const float*)d_in[1];
    const float* Wq   = (const float*)d_in[2];
    const float* bq   = (const float*)d_in[3];
    const float* Wk   = (const float*)d_in[4];
    const float* bk   = (const float*)d_in[5];
    const float* Wv   = (const float*)d_in[6];
    const float* bv   = (const float*)d_in[7];
    const float* temp = (const float*)d_in[8];

    float* attnOut = (float*)d_out;            // [8,4,48,48]
    float* vOut    = (float*)d_out + ATTN_N;   // [8,192,128,128]

    float* sAcc = (float*)d_ws;                // 73728 f32
    float* qss  = sAcc + ATTN_N;               // 1536  f32
    float* kss  = qss + 8 * CCH;               // 1536  f32
    const int zn = ATTN_N + 2 * 8 * CCH;       // ~307 KB of workspace used

    // atomics need zeroed accumulators every call (determinism)
    stmua_zero_kernel<<<(zn + 255) / 256, 256, 0, stream>>>((float*)d_ws, zn);

    const int nblocks = 8 * (HW / TILE_P);                // 1024
    const size_t ldsBytes = 3u * CCH * TILE_P * sizeof(float);  // 288 KB
    stmua_stage1<<<nblocks, 256, ldsBytes, stream>>>(
        inp1, inp2, Wq, bq, Wk, bk, Wv, bv, vOut, sAcc, qss, kss);

    stmua_softmax<<<8 * NHEAD, 64, 0, stream>>>(sAcc, qss, kss, temp, attnOut);
}